// DualVisionTokenExtractor_15479062135566
// MI455X (gfx1250) — compile-verified
//
#include <hip/hip_runtime.h>
#include <hip/hip_bf16.h>

#define BATCH 32
#define NPROP 1000
#define HDIM  4096
#define NLAB  92
#define TOPK  10
#define IOU_THR   0.7f
#define SCORE_THR 0.2f
#define IMGSZ 800.0f
#define NROWS (BATCH*TOPK)   /* 320 */
#define RTILES (NROWS/16)    /* 20  */
#define CTILES (HDIM/16)     /* 256 */

typedef __attribute__((ext_vector_type(2))) float v2f;
typedef __attribute__((ext_vector_type(8))) float v8f;

// ---------------------------------------------------------------- score sort (desc)
__global__ void k_sort(const float* __restrict__ boxes, const float* __restrict__ scores,
                       const int* __restrict__ labels,
                       float* __restrict__ sb, float* __restrict__ ss, int* __restrict__ sl) {
    int img = blockIdx.x;
    __shared__ float s_sc[NPROP];
    for (int i = threadIdx.x; i < NPROP; i += blockDim.x) s_sc[i] = scores[img*NPROP + i];
    __syncthreads();
    for (int i = threadIdx.x; i < NPROP; i += blockDim.x) {
        float si = s_sc[i];
        int rank = 0;
        for (int j = 0; j < NPROP; ++j) {
            float sj = s_sc[j];
            rank += (sj > si) || (sj == si && j < i);   // descending, stable
        }
        ss[img*NPROP + rank] = si;
        sl[img*NPROP + rank] = labels[img*NPROP + i];
        const float* bs = boxes + (size_t)(img*NPROP + i)*4;
        float* bd = sb + (size_t)(img*NPROP + rank)*4;
        bd[0] = bs[0]; bd[1] = bs[1]; bd[2] = bs[2]; bd[3] = bs[3];
    }
}

// ---------------------------------------------------------------- greedy NMS + compact
__global__ void k_nms(const float* __restrict__ sb, const float* __restrict__ ss,
                      const int* __restrict__ sl,
                      float* __restrict__ rowBox, float* __restrict__ rowScore,
                      int* __restrict__ rowLabel, int* __restrict__ rowMask) {
    int img = blockIdx.x;
    __shared__ float bx[NPROP*4];
    __shared__ float area[NPROP];
    __shared__ int   keep[NPROP];
    for (int i = threadIdx.x; i < NPROP; i += blockDim.x) {
        float x1 = sb[(size_t)(img*NPROP+i)*4+0];
        float y1 = sb[(size_t)(img*NPROP+i)*4+1];
        float x2 = sb[(size_t)(img*NPROP+i)*4+2];
        float y2 = sb[(size_t)(img*NPROP+i)*4+3];
        bx[i*4+0]=x1; bx[i*4+1]=y1; bx[i*4+2]=x2; bx[i*4+3]=y2;
        area[i] = fmaxf(x2-x1, 0.0f) * fmaxf(y2-y1, 0.0f);
        keep[i] = (ss[img*NPROP+i] >= SCORE_THR) ? 1 : 0;
    }
    __syncthreads();
    for (int i = 0; i < NPROP-1; ++i) {
        if (keep[i]) {
            float xi1=bx[i*4+0], yi1=bx[i*4+1], xi2=bx[i*4+2], yi2=bx[i*4+3];
            float ai = area[i];
            for (int j = i+1+threadIdx.x; j < NPROP; j += blockDim.x) {
                float lx = fmaxf(xi1, bx[j*4+0]);
                float ly = fmaxf(yi1, bx[j*4+1]);
                float rx = fminf(xi2, bx[j*4+2]);
                float ry = fminf(yi2, bx[j*4+3]);
                float iw = fmaxf(rx-lx, 0.0f), ih = fmaxf(ry-ly, 0.0f);
                float inter = iw*ih;
                float iou = inter / (ai + area[j] - inter + 1e-9f);
                if (iou > IOU_THR) keep[j] = 0;
            }
        }
        __syncthreads();
    }
    if (threadIdx.x == 0) {
        int cnt = 0;
        for (int i = 0; i < NPROP && cnt < TOPK; ++i) {
            if (!keep[i]) continue;
            float x1 = fminf(fmaxf(rintf(bx[i*4+0]), 0.0f), IMGSZ-1.0f);
            float y1 = fminf(fmaxf(rintf(bx[i*4+1]), 0.0f), IMGSZ-1.0f);
            float x2 = fminf(fmaxf(rintf(bx[i*4+2]), 0.0f), IMGSZ);
            float y2 = fminf(fmaxf(rintf(bx[i*4+3]), 0.0f), IMGSZ);
            if (!((x2 > x1 + 1.0f) && (y2 > y1 + 1.0f))) continue;
            int r = img*TOPK + cnt;
            rowBox[r*4+0] = x1/IMGSZ; rowBox[r*4+1] = y1/IMGSZ;
            rowBox[r*4+2] = x2/IMGSZ; rowBox[r*4+3] = y2/IMGSZ;
            rowScore[r] = ss[img*NPROP+i];
            int lb = sl[img*NPROP+i];
            rowLabel[r] = lb < 0 ? 0 : (lb > NLAB-1 ? NLAB-1 : lb);
            rowMask[r] = 1;
            ++cnt;
        }
        for (; cnt < TOPK; ++cnt) {
            int r = img*TOPK + cnt;
            rowBox[r*4+0]=0.0f; rowBox[r*4+1]=0.0f; rowBox[r*4+2]=0.0f; rowBox[r*4+3]=0.0f;
            rowScore[r]=0.0f; rowLabel[r]=0; rowMask[r]=0;
        }
    }
}

// ---------------------------------------------------------------- fused WMMA projection + box LN + det LN
// Each block handles one 16-row tile (16 rows x 4096 cols). 8 waves x 32 col tiles.
// The K=4 WMMA is so cheap we recompute it per phase instead of spilling tiles to
// HBM or LDS; LN statistics cross waves via ds_add_f32 LDS atomics.
__global__ void k_fused(const float* __restrict__ rowBox, const float* __restrict__ box_w,
                        const float* __restrict__ box_b,
                        const float* __restrict__ rowScore, const int* __restrict__ rowLabel,
                        const int* __restrict__ rowMask,
                        const float* __restrict__ label_emb, const float* __restrict__ score_w,
                        const float* __restrict__ score_b,
                        const float* __restrict__ bln_g, const float* __restrict__ bln_b,
                        const float* __restrict__ dln_g, const float* __restrict__ dln_b,
                        float* __restrict__ out_tok, int* __restrict__ out_mask) {
    __shared__ float sS1[16], sQ1[16], sS2[16], sQ2[16];
    __shared__ float sSc[16];
    __shared__ int   sLab[16], sMsk[16];

    const int rt    = blockIdx.x;          // 0..19
    const int wave  = threadIdx.x >> 5;    // 0..7
    const int lane  = threadIdx.x & 31;
    const int m     = lane & 15;
    const int k0    = (lane < 16) ? 0 : 2; // ISA A layout: lanes 0-15 -> K=0,1 ; 16-31 -> K=2,3
    const int lrb   = (lane < 16) ? 0 : 8; // C/D layout: vgpr r -> local row r (+8 hi half-wave)
    const float invH = 1.0f / (float)HDIM;

    if (threadIdx.x < 16) {
        sS1[threadIdx.x] = 0.0f; sQ1[threadIdx.x] = 0.0f;
        sS2[threadIdx.x] = 0.0f; sQ2[threadIdx.x] = 0.0f;
        int gr = rt*16 + threadIdx.x;
        sSc[threadIdx.x]  = rowScore[gr];
        sLab[threadIdx.x] = rowLabel[gr];
        sMsk[threadIdx.x] = rowMask[gr];
    }
    const int rowA = rt*16 + m;
    v2f a;
    a.x = rowBox[rowA*4 + k0];
    a.y = rowBox[rowA*4 + k0 + 1];
    __syncthreads();

    // ---- phase 1: bp_pre = box @ box_w + box_b ; accumulate box-LN stats
    float ps[8], pq[8];
#pragma unroll
    for (int r = 0; r < 8; ++r) { ps[r] = 0.0f; pq[r] = 0.0f; }
    for (int t = 0; t < CTILES/8; ++t) {
        int col = (wave + t*8)*16 + m;
        v2f bvec;
        bvec.x = box_w[(size_t)k0*HDIM + col];
        bvec.y = box_w[(size_t)(k0+1)*HDIM + col];
        float cb = box_b[col];
        v8f c = {cb, cb, cb, cb, cb, cb, cb, cb};
        v8f acc = __builtin_amdgcn_wmma_f32_16x16x4_f32(false, a, false, bvec,
                                                        (short)0, c, false, false);
#pragma unroll
        for (int r = 0; r < 8; ++r) { float v = acc[r]; ps[r] += v; pq[r] += v*v; }
    }
#pragma unroll
    for (int r = 0; r < 8; ++r) {
        atomicAdd(&sS1[lrb + r], ps[r]);    // ds_add_f32
        atomicAdd(&sQ1[lrb + r], pq[r]);
    }
    __syncthreads();

    float mu1[8], rs1[8];
#pragma unroll
    for (int r = 0; r < 8; ++r) {
        int lr = lrb + r;
        mu1[r] = sS1[lr] * invH;
        float var = sQ1[lr] * invH - mu1[r]*mu1[r];
        rs1[r] = rsqrtf(fmaxf(var, 0.0f) + 1e-5f);
    }

    // ---- phase 2: pre = label_emb + s*score_w + score_b + boxLN(bp_pre) ; det-LN stats
#pragma unroll
    for (int r = 0; r < 8; ++r) { ps[r] = 0.0f; pq[r] = 0.0f; }
    for (int t = 0; t < CTILES/8; ++t) {
        int col = (wave + t*8)*16 + m;
        v2f bvec;
        bvec.x = box_w[(size_t)k0*HDIM + col];
        bvec.y = box_w[(size_t)(k0+1)*HDIM + col];
        float cb = box_b[col];
        v8f c = {cb, cb, cb, cb, cb, cb, cb, cb};
        v8f acc = __builtin_amdgcn_wmma_f32_16x16x4_f32(false, a, false, bvec,
                                                        (short)0, c, false, false);
        float g = bln_g[col], bb = bln_b[col];
        float sw = score_w[col], sbv = score_b[col];
#pragma unroll
        for (int r = 0; r < 8; ++r) {
            int lr = lrb + r;
            float bpn = (acc[r] - mu1[r]) * rs1[r] * g + bb;
            float pre = label_emb[(size_t)sLab[lr]*HDIM + col] + sSc[lr]*sw + sbv + bpn;
            ps[r] += pre; pq[r] += pre*pre;
        }
    }
#pragma unroll
    for (int r = 0; r < 8; ++r) {
        atomicAdd(&sS2[lrb + r], ps[r]);
        atomicAdd(&sQ2[lrb + r], pq[r]);
    }
    __syncthreads();

    float mu2[8], rs2[8];
#pragma unroll
    for (int r = 0; r < 8; ++r) {
        int lr = lrb + r;
        mu2[r] = sS2[lr] * invH;
        float var = sQ2[lr] * invH - mu2[r]*mu2[r];
        rs2[r] = rsqrtf(fmaxf(var, 0.0f) + 1e-5f);
    }

    // ---- phase 3: recompute pre, apply det LN, mask, write tokens
    for (int t = 0; t < CTILES/8; ++t) {
        int col = (wave + t*8)*16 + m;
        v2f bvec;
        bvec.x = box_w[(size_t)k0*HDIM + col];
        bvec.y = box_w[(size_t)(k0+1)*HDIM + col];
        float cb = box_b[col];
        v8f c = {cb, cb, cb, cb, cb, cb, cb, cb};
        v8f acc = __builtin_amdgcn_wmma_f32_16x16x4_f32(false, a, false, bvec,
                                                        (short)0, c, false, false);
        float g = bln_g[col], bb = bln_b[col];
        float sw = score_w[col], sbv = score_b[col];
        float dg = dln_g[col], db = dln_b[col];
#pragma unroll
        for (int r = 0; r < 8; ++r) {
            int lr = lrb + r;
            float bpn = (acc[r] - mu1[r]) * rs1[r] * g + bb;
            float pre = label_emb[(size_t)sLab[lr]*HDIM + col] + sSc[lr]*sw + sbv + bpn;
            float tok = (pre - mu2[r]) * rs2[r] * dg + db;
            out_tok[(size_t)(rt*16 + lr)*HDIM + col] = sMsk[lr] ? tok : 0.0f;
        }
    }
    if (threadIdx.x < 16) out_mask[rt*16 + threadIdx.x] = sMsk[threadIdx.x];
}

// ---------------------------------------------------------------- launch
extern "C" void kernel_launch(void* const* d_in, const int* in_sizes, int n_in,
                              void* d_out, int out_size, void* d_ws, size_t ws_size,
                              hipStream_t stream) {
    const float* boxes     = (const float*)d_in[0];
    const float* scores    = (const float*)d_in[1];
    const int*   labels    = (const int*)  d_in[2];
    const float* label_emb = (const float*)d_in[3];
    const float* score_w   = (const float*)d_in[4];
    const float* score_b   = (const float*)d_in[5];
    const float* box_w     = (const float*)d_in[6];
    const float* box_b     = (const float*)d_in[7];
    const float* bln_g     = (const float*)d_in[8];
    const float* bln_b     = (const float*)d_in[9];
    const float* dln_g     = (const float*)d_in[10];
    const float* dln_b     = (const float*)d_in[11];

    char* ws = (char*)d_ws;
    size_t off = 0;
    auto carve = [&](size_t bytes) -> void* {
        void* p = ws + off;
        off = (off + bytes + 255) & ~(size_t)255;
        return p;
    };
    float* sb       = (float*)carve(sizeof(float) * BATCH * NPROP * 4);
    float* ss       = (float*)carve(sizeof(float) * BATCH * NPROP);
    int*   sl       = (int*)  carve(sizeof(int)   * BATCH * NPROP);
    float* rowBox   = (float*)carve(sizeof(float) * NROWS * 4);
    float* rowScore = (float*)carve(sizeof(float) * NROWS);
    int*   rowLabel = (int*)  carve(sizeof(int)   * NROWS);
    int*   rowMask  = (int*)  carve(sizeof(int)   * NROWS);

    float* out_tok = (float*)d_out;
    int*   out_msk = (int*)(out_tok + (size_t)NROWS * HDIM);

    k_sort <<<BATCH, 256, 0, stream>>>(boxes, scores, labels, sb, ss, sl);
    k_nms  <<<BATCH, 256, 0, stream>>>(sb, ss, sl, rowBox, rowScore, rowLabel, rowMask);
    k_fused<<<RTILES, 256, 0, stream>>>(rowBox, box_w, box_b, rowScore, rowLabel, rowMask,
                                        label_emb, score_w, score_b,
                                        bln_g, bln_b, dln_g, dln_b, out_tok, out_msk);
}